// BoundaryAwarePooling_13529146983189
// MI455X (gfx1250) — compile-verified
//
#include <hip/hip_runtime.h>
#include <hip/hip_bf16.h>

typedef __attribute__((ext_vector_type(2))) float v2f;
typedef __attribute__((ext_vector_type(8))) float v8f;

#define B_      16
#define T_      8192
#define D_      256
#define MAXSEG  4608   // >= worst case 4096 segments per row
#define LN_EPS  1e-5f

// ---------------------------------------------------------------------------
// Kernel 1: empty-row value = LayerNorm(b_proj) (what the reference produces
// for rows whose segment is empty: pooled=0 -> out=b_proj -> LN).
// ---------------------------------------------------------------------------
__global__ __launch_bounds__(256) void k_ln_bias(const float* __restrict__ bproj,
                                                 const float* __restrict__ gamma,
                                                 const float* __restrict__ beta,
                                                 float* __restrict__ emptyrow) {
    __shared__ float s1[256], s2[256];
    int t = threadIdx.x;
    float v = bproj[t];
    s1[t] = v; s2[t] = v * v;
    __syncthreads();
    for (int o = 128; o > 0; o >>= 1) {
        if (t < o) { s1[t] += s1[t + o]; s2[t] += s2[t + o]; }
        __syncthreads();
    }
    float mu  = s1[0] * (1.0f / D_);
    float var = s2[0] * (1.0f / D_) - mu * mu;
    float r   = rsqrtf(var + LN_EPS);
    emptyrow[t] = (v - mu) * r * gamma[t] + beta[t];
}

// ---------------------------------------------------------------------------
// Kernel 2: pack W into K-pair float2s: Wp[k2][e] = {W[e][2k2], W[e][2k2+1]}.
// A WMMA-B fragment (K=ka,ka+1 per lane) then loads as one global_load_b64.
// Block = output row e (coalesced float2 read of W[e][*]); one-time 256 KB.
// ---------------------------------------------------------------------------
__global__ __launch_bounds__(128) void k_pack_w(const float* __restrict__ W,
                                                float2* __restrict__ Wp) {
    int e  = blockIdx.x;        // 256
    int k2 = threadIdx.x;       // 128 K-pairs
    float2 v = *(const float2*)(W + (size_t)e * D_ + 2 * k2);
    Wp[(size_t)k2 * D_ + e] = v;
}

// ---------------------------------------------------------------------------
// Kernel 3: fill all B*T output rows with the empty-row value.
// ---------------------------------------------------------------------------
__global__ __launch_bounds__(256) void k_fill(const float* __restrict__ emptyrow,
                                              float* __restrict__ out) {
    int t = threadIdx.x;
    float v = emptyrow[t];
    size_t row0 = (size_t)blockIdx.x * 16;
    float* p = out + row0 * D_ + t;
#pragma unroll
    for (int r = 0; r < 16; ++r) p[(size_t)r * D_] = v;
}

// ---------------------------------------------------------------------------
// Kernel 4: per-row segment extraction. One block (1024 thr = 32 waves) per
// batch row. Segments are contiguous runs of non-boundary tokens; the k-th
// start pairs with the k-th end, so two ordered prefix-scans give [start,end).
// ---------------------------------------------------------------------------
__global__ __launch_bounds__(1024) void k_scan(const long long* __restrict__ ids,
                                               int* __restrict__ starts,
                                               int* __restrict__ ends,
                                               int* __restrict__ nseg) {
    int b = blockIdx.x;
    const long long* row = ids + (size_t)b * T_;
    __shared__ int wsum_s[32], wsum_e[32];
    __shared__ int woff_s[32], woff_e[32];
    __shared__ int base_s, base_e;
    if (threadIdx.x == 0) { base_s = 0; base_e = 0; }
    __syncthreads();
    int wave = threadIdx.x >> 5;
    int lane = threadIdx.x & 31;
    unsigned long long lmask = (1ull << lane) - 1ull;

    for (int c0 = 0; c0 < T_; c0 += 1024) {
        int t = c0 + threadIdx.x;
        bool nb    = (row[t] != 0);
        bool prevb = (t == 0)      ? true : (row[t - 1] == 0);
        bool nextb = (t == T_ - 1) ? true : (row[t + 1] == 0);
        bool sf = nb && prevb;   // segment start
        bool ef = nb && nextb;   // segment end (last token of run)
        unsigned long long ms = __ballot(sf);
        unsigned long long me = __ballot(ef);
        int ps = __popcll(ms & lmask);
        int pe = __popcll(me & lmask);
        if (lane == 0) { wsum_s[wave] = __popcll(ms); wsum_e[wave] = __popcll(me); }
        __syncthreads();
        if (threadIdx.x == 0) {
            int as = base_s, ae = base_e;
            for (int i = 0; i < 32; ++i) {
                woff_s[i] = as; as += wsum_s[i];
                woff_e[i] = ae; ae += wsum_e[i];
            }
            base_s = as; base_e = ae;
        }
        __syncthreads();
        if (sf) starts[b * MAXSEG + woff_s[wave] + ps] = t;
        if (ef) ends  [b * MAXSEG + woff_e[wave] + pe] = t + 1;
        __syncthreads();
    }
    if (threadIdx.x == 0) nseg[b] = base_s;
}

// ---------------------------------------------------------------------------
// Kernel 5: fused pool -> GEMM (v_wmma_f32_16x16x4_f32) -> LayerNorm.
// Block = 512 thr (16 waves) handles a 16-segment tile of one batch row.
//   Phase A: wave i mean-pools segment w0+i into LDS (16x256 f32).
//   Phase B: wave n computes output columns [16n,16n+16) via 64 WMMA k-steps:
//       A frag (16x4 f32, ISA layout): lanes 0-15 hold M, VGPR0=K=ka,
//       VGPR1=K=ka+1 with ka = k + 2*half; B frag mirrors with N across
//       lanes, loaded as one b64 from the K-pair-packed Wp.
//   Phase C: bias add into LDS, per-row LN via wave shuffle reduce, store.
// ---------------------------------------------------------------------------
__global__ __launch_bounds__(512) void k_pool_gemm_ln(
    const float* __restrict__ x, const float2* __restrict__ Wp,
    const float* __restrict__ bproj, const float* __restrict__ gamma,
    const float* __restrict__ beta, const int* __restrict__ starts,
    const int* __restrict__ ends, const int* __restrict__ nseg,
    float* __restrict__ out) {

    __shared__ float tile[16][D_];   // pooled rows, later reused for out tile

    int b  = blockIdx.y;
    int ns = nseg[b];
    int w0 = blockIdx.x * 16;
    if (w0 >= ns) return;            // block-uniform: EXEC stays full below

    int wave = threadIdx.x >> 5;
    int lane = threadIdx.x & 31;
    int half = lane >> 4;            // 0: lanes 0-15, 1: lanes 16-31
    int lm   = lane & 15;

    // -------- Phase A: pooling (wave-uniform control flow) ----------------
    int w = w0 + wave;
    float a0 = 0.f, a1 = 0.f, a2 = 0.f, a3 = 0.f;
    float a4 = 0.f, a5 = 0.f, a6 = 0.f, a7 = 0.f;
    float inv = 0.f;
    if (w < ns) {
        int s = starts[b * MAXSEG + w];
        int e = ends  [b * MAXSEG + w];
        inv = 1.0f / (float)(e - s);
        const float* xr = x + ((size_t)b * T_ + s) * D_;
        for (int t = s; t < e; ++t, xr += D_) {
            __builtin_prefetch(xr + 2 * D_, 0, 1);          // global_prefetch_b8
            float4 v0 = *(const float4*)(xr + lane * 4);
            float4 v1 = *(const float4*)(xr + 128 + lane * 4);
            a0 += v0.x; a1 += v0.y; a2 += v0.z; a3 += v0.w;
            a4 += v1.x; a5 += v1.y; a6 += v1.z; a7 += v1.w;
        }
    }
    float4 p0 = { a0 * inv, a1 * inv, a2 * inv, a3 * inv };
    float4 p1 = { a4 * inv, a5 * inv, a6 * inv, a7 * inv };
    *(float4*)&tile[wave][lane * 4]       = p0;   // zeros for w >= ns
    *(float4*)&tile[wave][128 + lane * 4] = p1;
    __syncthreads();

    // -------- Phase B: GEMM, wave handles output cols [e0, e0+16) ---------
    int e0 = wave * 16;
    // B pointer: K-pair index (k/2 + half), lane picks column e0+lm.
    const float2* wq = Wp + (size_t)half * D_ + e0 + lm;
    v8f c = (v8f)0.0f;
#pragma unroll 8
    for (int k = 0; k < D_; k += 4) {
        int ka = k + 2 * half;
        v2f A  = *(const v2f*)&tile[lm][ka];     // K=ka, ka+1 (ds_load b64)
        v2f Bv = *(const v2f*)wq;                // K=ka, ka+1 (global b64)
        wq += 2 * D_;
        c = __builtin_amdgcn_wmma_f32_16x16x4_f32(
                false, A, false, Bv, (short)0, c, false, false);
    }
    __syncthreads();   // everyone done reading pooled rows

    // -------- Phase C: bias into LDS, then LayerNorm per row --------------
#pragma unroll
    for (int r = 0; r < 8; ++r) {
        int M   = r + 8 * half;
        int col = e0 + lm;
        tile[M][col] = c[r] + bproj[col];
    }
    __syncthreads();

    if (w < ns) {      // wave-uniform; wave normalizes/stores row `wave`
        float4 u0 = *(const float4*)&tile[wave][lane * 4];
        float4 u1 = *(const float4*)&tile[wave][128 + lane * 4];
        float s  = u0.x + u0.y + u0.z + u0.w + u1.x + u1.y + u1.z + u1.w;
        float sq = u0.x*u0.x + u0.y*u0.y + u0.z*u0.z + u0.w*u0.w +
                   u1.x*u1.x + u1.y*u1.y + u1.z*u1.z + u1.w*u1.w;
        for (int off = 16; off > 0; off >>= 1) {
            s  += __shfl_xor(s,  off);
            sq += __shfl_xor(sq, off);
        }
        float mu  = s * (1.0f / D_);
        float var = sq * (1.0f / D_) - mu * mu;
        float rn  = rsqrtf(var + LN_EPS);

        float4 g0 = *(const float4*)(gamma + lane * 4);
        float4 g1 = *(const float4*)(gamma + 128 + lane * 4);
        float4 t0 = *(const float4*)(beta + lane * 4);
        float4 t1 = *(const float4*)(beta + 128 + lane * 4);
        float4 o0, o1;
        o0.x = (u0.x - mu) * rn * g0.x + t0.x;
        o0.y = (u0.y - mu) * rn * g0.y + t0.y;
        o0.z = (u0.z - mu) * rn * g0.z + t0.z;
        o0.w = (u0.w - mu) * rn * g0.w + t0.w;
        o1.x = (u1.x - mu) * rn * g1.x + t1.x;
        o1.y = (u1.y - mu) * rn * g1.y + t1.y;
        o1.z = (u1.z - mu) * rn * g1.z + t1.z;
        o1.w = (u1.w - mu) * rn * g1.w + t1.w;
        float* orow = out + ((size_t)b * T_ + w) * D_;
        *(float4*)(orow + lane * 4)       = o0;
        *(float4*)(orow + 128 + lane * 4) = o1;
    }
}

// ---------------------------------------------------------------------------
// Kernel 6: mask[b,w] = (w < nseg[b])
// ---------------------------------------------------------------------------
__global__ __launch_bounds__(256) void k_mask(const int* __restrict__ nseg,
                                              float* __restrict__ mask) {
    int idx = blockIdx.x * 256 + threadIdx.x;
    int b = idx / T_;
    int w = idx - b * T_;
    mask[idx] = (w < nseg[b]) ? 1.0f : 0.0f;
}

// ---------------------------------------------------------------------------
extern "C" void kernel_launch(void* const* d_in, const int* in_sizes, int n_in,
                              void* d_out, int out_size, void* d_ws, size_t ws_size,
                              hipStream_t stream) {
    const float*     x     = (const float*)d_in[0];
    const long long* ids   = (const long long*)d_in[1];
    const float*     W     = (const float*)d_in[2];
    const float*     bproj = (const float*)d_in[3];
    const float*     gamma = (const float*)d_in[4];
    const float*     beta  = (const float*)d_in[5];

    float* out  = (float*)d_out;                         // [B, T, D]
    float* mask = out + (size_t)B_ * T_ * D_;            // [B, T]

    char* ws = (char*)d_ws;
    size_t off = 0;
    int*    starts   = (int*)(ws + off);    off += (size_t)B_ * MAXSEG * sizeof(int);
    int*    ends     = (int*)(ws + off);    off += (size_t)B_ * MAXSEG * sizeof(int);
    int*    nseg     = (int*)(ws + off);    off += 256;   // 16 ints, padded
    float2* Wpk      = (float2*)(ws + off); off += (size_t)D_ * D_ * sizeof(float);
    float*  emptyrow = (float*)(ws + off);  off += D_ * sizeof(float);
    (void)ws_size; (void)in_sizes; (void)n_in; (void)out_size;

    k_ln_bias<<<1, 256, 0, stream>>>(bproj, gamma, beta, emptyrow);
    k_pack_w<<<D_, 128, 0, stream>>>(W, Wpk);
    k_fill<<<(B_ * T_) / 16, 256, 0, stream>>>(emptyrow, out);
    k_scan<<<B_, 1024, 0, stream>>>(ids, starts, ends, nseg);
    dim3 grid(MAXSEG / 16, B_);
    k_pool_gemm_ln<<<grid, 512, 0, stream>>>(x, Wpk, bproj, gamma, beta,
                                             starts, ends, nseg, out);
    k_mask<<<(B_ * T_) / 256, 256, 0, stream>>>(nseg, mask);
}